// BiGNN_50663434224368
// MI455X (gfx1250) — compile-verified
//
#include <hip/hip_runtime.h>

typedef float v2f __attribute__((ext_vector_type(2)));
typedef float v8f __attribute__((ext_vector_type(8)));

#define N_NODES_C 50000
#define N_EDGES_C 1600000

// ---------------------------------------------------------------------------
// Edge kernel: fused  edge-MLP (4->25->128 via WMMA f32)  +  per-edge matmul
//              +  atomic scatter-add to agg[dst].
// One wave (32 lanes) processes tiles of 16 edges.
//   A-matrix (16 edges x 28) built directly in WMMA A layout:
//     lane l: m = l&15, khalf = 2*(l>>4); chunk c holds K = 4c+khalf, 4c+1+khalf
//   B-matrix = wb (25x128, zero-padded to 28 rows) staged in LDS, read in
//     B layout: VGPR0 row 4c+khalf, VGPR1 row 4c+1+khalf, col = 16t + (l&15)
//   D layout: lane l, vgpr r  ->  edge m = r + 8*(l>>4), col n = 16t + (l&15)
//   n = i*OUTC + o  with  o = (l&15) % OUTC  (t-invariant)  and i = n / OUTC.
// ---------------------------------------------------------------------------
template <int IN, int OUTC>
__global__ __launch_bounds__(256) void edge_kernel(
    const float* __restrict__ x,          // [N, IN]
    const int* __restrict__ src,          // [E]
    const int* __restrict__ dst,          // [E]
    const float* __restrict__ ea,         // [E, 4]
    const float* __restrict__ wa,         // [4, 25]
    const float* __restrict__ ba,         // [25]
    const float* __restrict__ wb,         // [25, 128]
    const float* __restrict__ bb,         // [128]
    float* __restrict__ agg,              // [N, OUTC]  (pre-zeroed)
    int nEdges)
{
    __shared__ float s_wb[28 * 132];      // zero-padded, stride 132
    __shared__ float s_wa[4 * 28];        // zero-padded
    __shared__ float s_ba[28];
    __shared__ float s_bb[128];

    const int tid = threadIdx.x;

    // ---- stage constants into LDS (zero-pad, then fill) ----
    for (int i = tid; i < 28 * 132; i += 256) s_wb[i] = 0.f;
    for (int i = tid; i < 4 * 28; i += 256) s_wa[i] = 0.f;
    if (tid < 28) s_ba[tid] = 0.f;
    __syncthreads();
    for (int i = tid; i < 25 * 128; i += 256) {
        int r = i >> 7, c = i & 127;
        s_wb[r * 132 + c] = wb[i];
    }
    for (int i = tid; i < 4 * 25; i += 256) {
        int j = i / 25, k = i - j * 25;
        s_wa[j * 28 + k] = wa[i];
    }
    if (tid < 25) s_ba[tid] = ba[tid];
    if (tid < 128) s_bb[tid] = bb[tid];
    __syncthreads();

    const int lane  = tid & 31;
    const int m     = lane & 15;          // edge row this lane stages
    const int half  = lane >> 4;          // 0 | 1
    const int khalf = half * 2;
    const int colL  = lane & 15;          // column within a 16-wide N tile
    const int o_ch  = colL % OUTC;        // output channel (t-invariant)

    const int gwave  = blockIdx.x * 8 + (tid >> 5);
    const int nWaves = gridDim.x * 8;
    const int nTiles = (nEdges + 15) >> 4;

    for (int tile = gwave; tile < nTiles; tile += nWaves) {
        const int base = tile << 4;
        const int e_m  = base + m;
        const bool valid = (e_m < nEdges);

        int sNode = 0;
        int dstr  = -1;
        float4 ev = make_float4(0.f, 0.f, 0.f, 0.f);
        if (valid) {
            sNode = src[e_m];
            dstr  = dst[e_m];
            ev    = reinterpret_cast<const float4*>(ea)[e_m];
        }

        // gather x[src] row into registers (both halves hold edge m's row)
        float xr[IN];
#pragma unroll
        for (int q = 0; q < IN; ++q) xr[q] = 0.f;
        if (valid) {
            const float4* xp = reinterpret_cast<const float4*>(x + (size_t)sNode * IN);
#pragma unroll
            for (int q = 0; q < IN / 4; ++q) {
                float4 v = xp[q];
                xr[4 * q + 0] = v.x; xr[4 * q + 1] = v.y;
                xr[4 * q + 2] = v.z; xr[4 * q + 3] = v.w;
            }
        }

        // ---- first MLP layer, computed directly in WMMA A layout ----
        // h1[m,k] = relu(ba[k] + sum_j ea[m,j]*wa[j,k]); padded k>=25 -> 0
        v2f A[7];
#pragma unroll
        for (int c = 0; c < 7; ++c) {
            const int k0 = 4 * c + khalf;
            float h0 = s_ba[k0] + ev.x * s_wa[k0] + ev.y * s_wa[28 + k0]
                     + ev.z * s_wa[56 + k0] + ev.w * s_wa[84 + k0];
            float h1 = s_ba[k0 + 1] + ev.x * s_wa[k0 + 1] + ev.y * s_wa[28 + k0 + 1]
                     + ev.z * s_wa[56 + k0 + 1] + ev.w * s_wa[84 + k0 + 1];
            A[c].x = fmaxf(h0, 0.f);
            A[c].y = fmaxf(h1, 0.f);
        }

        // ---- big GEMM h1[16x28] @ wb[28x128] via WMMA, fused consume ----
        float msg[8];
#pragma unroll
        for (int r = 0; r < 8; ++r) msg[r] = 0.f;

#pragma unroll
        for (int t = 0; t < 8; ++t) {
            const float bbv = s_bb[16 * t + colL];   // bias folded into C
            v8f acc = {bbv, bbv, bbv, bbv, bbv, bbv, bbv, bbv};
#pragma unroll
            for (int c = 0; c < 7; ++c) {
                v2f b;
                const int bi = (4 * c + khalf) * 132 + 16 * t + colL;
                b.x = s_wb[bi];
                b.y = s_wb[bi + 132];
                acc = __builtin_amdgcn_wmma_f32_16x16x4_f32(
                    false, A[c], false, b, (short)0, acc, false, false);
            }
            // consume tile t into msg: msg[m,o] += x[m, n/OUTC] * W[m, n]
#pragma unroll
            for (int r = 0; r < 8; ++r) {
                const int srcl = r + 8 * half;       // lane holding edge m's data
                float xv;
                if constexpr (OUTC == 16) {          // IN = 8, i = t
                    xv = __shfl(xr[t], srcl, 32);
                } else {                             // IN = 16, i = 2t + (colL>=8)
                    float x0 = __shfl(xr[2 * t + 0], srcl, 32);
                    float x1 = __shfl(xr[2 * t + 1], srcl, 32);
                    xv = (colL >= 8) ? x1 : x0;
                }
                msg[r] = fmaf(xv, acc[r], msg[r]);
            }
        }

        // ---- scatter-add messages to agg[dst] ----
#pragma unroll
        for (int r = 0; r < 8; ++r) {
            const int srcl = r + 8 * half;
            const int d = __shfl(dstr, srcl, 32);
            if (d >= 0) atomicAdd(&agg[(size_t)d * OUTC + o_ch], msg[r]);
        }
    }
}

// ---------------------------------------------------------------------------
__global__ __launch_bounds__(256) void count_kernel(const int* __restrict__ dst,
                                                    float* __restrict__ cnt, int E)
{
    int e = blockIdx.x * 256 + threadIdx.x;
    if (e < E) atomicAdd(&cnt[dst[e]], 1.0f);
}

// out[n,o] = agg[n,o]/max(cnt,1) + sum_i x[n,i]*root[i,o] + bias[o]  (opt relu)
template <int IN, int OUTC, bool RELU>
__global__ __launch_bounds__(256) void node_kernel(
    const float* __restrict__ agg, const float* __restrict__ cnt,
    const float* __restrict__ x, const float* __restrict__ root,
    const float* __restrict__ bias, float* __restrict__ out, int nNodes)
{
    int idx = blockIdx.x * 256 + threadIdx.x;
    if (idx >= nNodes * OUTC) return;
    int n = idx / OUTC;
    int o = idx - n * OUTC;
    float c = fmaxf(cnt[n], 1.0f);
    float v = agg[idx] / c + bias[o];
#pragma unroll
    for (int i = 0; i < IN; ++i)
        v = fmaf(x[(size_t)n * IN + i], root[i * OUTC + o], v);
    if (RELU) v = fmaxf(v, 0.f);
    out[idx] = v;
}

// ---------------------------------------------------------------------------
extern "C" void kernel_launch(void* const* d_in, const int* in_sizes, int n_in,
                              void* d_out, int out_size, void* d_ws, size_t ws_size,
                              hipStream_t stream)
{
    const float* x     = (const float*)d_in[0];
    const int*   eidx  = (const int*)d_in[1];
    const float* ea    = (const float*)d_in[2];
    const float* w1a   = (const float*)d_in[3];
    const float* b1a   = (const float*)d_in[4];
    const float* w1b   = (const float*)d_in[5];
    const float* b1b   = (const float*)d_in[6];
    const float* root1 = (const float*)d_in[7];
    const float* bias1 = (const float*)d_in[8];
    const float* w2a   = (const float*)d_in[9];
    const float* b2a   = (const float*)d_in[10];
    const float* w2b   = (const float*)d_in[11];
    const float* b2b   = (const float*)d_in[12];
    const float* root2 = (const float*)d_in[13];
    const float* bias2 = (const float*)d_in[14];

    const int* src = eidx;
    const int* dst = eidx + N_EDGES_C;

    float* ws   = (float*)d_ws;
    float* agg1 = ws;                      // 50000*16
    float* h    = ws + 800000;             // 50000*16
    float* agg2 = ws + 1600000;            // 50000*8
    float* cnt  = ws + 2000000;            // 50000
    // zero agg1/h/agg2/cnt (h fully overwritten anyway) — capture-safe memset
    hipMemsetAsync(d_ws, 0, (size_t)2050000 * sizeof(float), stream);

    // degree counts (shared by both layers)
    count_kernel<<<(N_EDGES_C + 255) / 256, 256, 0, stream>>>(dst, cnt, N_EDGES_C);

    // layer 1: NNConv(8 -> 16) + relu
    edge_kernel<8, 16><<<1024, 256, 0, stream>>>(
        x, src, dst, ea, w1a, b1a, w1b, b1b, agg1, N_EDGES_C);
    node_kernel<8, 16, true><<<(N_NODES_C * 16 + 255) / 256, 256, 0, stream>>>(
        agg1, cnt, x, root1, bias1, h, N_NODES_C);

    // layer 2: NNConv(16 -> 8)
    edge_kernel<16, 8><<<1024, 256, 0, stream>>>(
        h, src, dst, ea, w2a, b2a, w2b, b2b, agg2, N_EDGES_C);
    node_kernel<16, 8, false><<<(N_NODES_C * 8 + 255) / 256, 256, 0, stream>>>(
        agg2, cnt, h, root2, bias2, (float*)d_out, N_NODES_C);
}